// QueryMaskDecoder_78228534329367
// MI455X (gfx1250) — compile-verified
//
#include <hip/hip_runtime.h>
#include <hip/hip_bf16.h>

// ---------------- problem constants (from reference) ----------------
#define BQ   16     // batch
#define CDIM 256    // channels
#define HW   1024   // H*W (32x32)
#define WMQ  128
#define HWM  16384  // Hm*Wm (128x128)
#define NQ   100    // queries
#define FDIM 1024   // FFN dim
#define NHH  8      // heads
#define HD   32     // head dim
#define LNUM 2      // layers
#define LN_EPS 1e-5f
#define MASK_NEG (-1e30f)

// GEMM tiling
#define KC    32            // K chunk (all K here are multiples of 32)
#define CTILE 128           // 8 waves * 16 columns
#define ASTR  (KC + 2)      // LDS row stride for A (pad: distinct banks per lane)
#define BSTR  (CTILE + 4)   // LDS row stride for B (pad: break half-wave conflicts)

typedef float v2f __attribute__((ext_vector_type(2)));
typedef float v4f __attribute__((ext_vector_type(4)));
typedef float v8f __attribute__((ext_vector_type(8)));

// ---------------- block-cooperative fp32 WMMA GEMM tile ----------------
// Block: 256 threads = 8 waves. Output tile: 16 rows x 128 cols (16/wave).
// A: row-major MxK (lda), OOB rows zeroed arithmetically (no divergence).
// B: row-major KxN (ldb), load columns clamped to [0, bcolmax).
// Per ISA 7.12.2 fp32 layouts:
//   A 16x4: lanes 0-15 K=0,1 ; lanes 16-31 K=2,3 (2 VGPRs)
//   B 4x16: mirrored: lanes 0-15 rows K=0,1 ; lanes 16-31 rows K=2,3
//   C/D: 8 VGPRs; lanes 0-15 rows r0..r0+7, lanes 16-31 rows r0+8..r0+15
__device__ __forceinline__ v8f gemm_block_tile(const float* __restrict__ A, int lda,
                                               int row0, int M,
                                               const float* __restrict__ Bm, int ldb,
                                               int bcol0, int bcolmax, int K,
                                               float* __restrict__ As,
                                               float* __restrict__ Bs)
{
    const int t    = threadIdx.x;       // 0..255
    const int lane = t & 31;
    const int half = lane >> 4;
    const int lm   = lane & 15;
    const int w    = t >> 5;            // wave id 0..7

    // A fill geometry: 16 rows x 32 cols, one b64 per thread
    const int ar = t >> 4;              // 0..15
    const int ac = (t & 15) * 2;        // 0..30
    const int grow  = row0 + ar;
    const float amask = (grow < M) ? 1.f : 0.f;
    const float* aptr = A + (size_t)(grow < M ? grow : 0) * lda;

    v8f acc = {0.f, 0.f, 0.f, 0.f, 0.f, 0.f, 0.f, 0.f};

    for (int k0 = 0; k0 < K; k0 += KC) {
        // ---- cooperative A fill (guard via multiply, never via branch) ----
        v2f av = *(const v2f*)(aptr + k0 + ac);
        As[ar * ASTR + ac]     = av[0] * amask;
        As[ar * ASTR + ac + 1] = av[1] * amask;

        // ---- cooperative B fill: 32 rows x 128 cols, four b128 per thread ----
        #pragma unroll
        for (int i = 0; i < 4; ++i) {
            const int fidx = t + i * 256;       // float4 index 0..1023
            const int br = fidx >> 5;           // row 0..31
            const int bc = (fidx & 31) * 4;     // col 0..124
            int gc = bcol0 + bc;
            gc = (gc > bcolmax - 4) ? (bcolmax - 4) : gc;   // clamp, stays aligned
            v4f bv = *(const v4f*)(Bm + (size_t)(k0 + br) * ldb + gc);
            Bs[br * BSTR + bc]     = bv[0];
            Bs[br * BSTR + bc + 1] = bv[1];
            Bs[br * BSTR + bc + 2] = bv[2];
            Bs[br * BSTR + bc + 3] = bv[3];
        }
        __syncthreads();

        // ---- 8 WMMAs per chunk from LDS ----
        const float* asr = &As[lm * ASTR + half * 2];
        const float* bsr = &Bs[(half * 2) * BSTR + w * 16 + lm];
        #pragma unroll
        for (int kk = 0; kk < KC; kk += 4) {
            v2f a = *(const v2f*)(asr + kk);
            v2f b;
            b[0] = bsr[(size_t)kk * BSTR];
            b[1] = bsr[(size_t)(kk + 1) * BSTR];
            acc = __builtin_amdgcn_wmma_f32_16x16x4_f32(false, a, false, b,
                                                        (short)0, acc, false, false);
        }
        __syncthreads();
    }
    return acc;
}

__device__ __forceinline__ void store_tile(float* __restrict__ Y, int ldy,
                                           int row0, int M, int col0, int Nc,
                                           const v8f& acc,
                                           const float* __restrict__ bias, int relu)
{
    if (col0 >= Nc) return;
    const int lane = threadIdx.x & 31;
    const int half = lane >> 4;
    const int lm   = lane & 15;
    const int c    = col0 + lm;
    const float bb = bias ? bias[c] : 0.f;
    #pragma unroll
    for (int j = 0; j < 8; ++j) {
        const int r = row0 + j + half * 8;
        if (r < M) {
            float v = acc[j] + bb;
            if (relu) v = fmaxf(v, 0.f);
            Y[(size_t)r * ldy + c] = v;
        }
    }
}

// ---------------- batched GEMM: Y[b] = act(X[b] @ W + bias) ----------------
__global__ void __launch_bounds__(256)
k_gemm_xw(const float* __restrict__ X, const float* __restrict__ W,
          const float* __restrict__ bias, float* __restrict__ Y,
          int M, int K, int Nc, int relu)
{
    __shared__ float As[16 * ASTR];
    __shared__ float Bs[KC * BSTR];
    const int b = blockIdx.z;
    const int row0  = blockIdx.x * 16;
    const int bcol0 = blockIdx.y * CTILE;
    const float* Xb = X + (size_t)b * M * K;
    float*       Yb = Y + (size_t)b * M * Nc;
    v8f acc = gemm_block_tile(Xb, K, row0, M, W, Nc, bcol0, Nc, K, As, Bs);
    const int col0 = bcol0 + (threadIdx.x >> 5) * 16;
    store_tile(Yb, Nc, row0, M, col0, Nc, acc, bias, relu);
}

// ---------------- mask einsum: out[b,n,p] = sum_c e[b,n,c] * mf[b,c,p] ----------------
__global__ void __launch_bounds__(256)
k_mask_einsum(const float* __restrict__ e, const float* __restrict__ mf,
              float* __restrict__ out)
{
    __shared__ float As[16 * ASTR];
    __shared__ float Bs[KC * BSTR];
    const int b = blockIdx.z;
    const int row0  = blockIdx.x * 16;
    const int bcol0 = blockIdx.y * CTILE;
    const float* Eb = e   + (size_t)b * NQ * CDIM;
    const float* Mb = mf  + (size_t)b * CDIM * HWM;
    float*       Ob = out + (size_t)b * NQ * HWM;
    v8f acc = gemm_block_tile(Eb, CDIM, row0, NQ, Mb, HWM, bcol0, HWM, CDIM, As, Bs);
    const int col0 = bcol0 + (threadIdx.x >> 5) * 16;
    store_tile(Ob, HWM, row0, NQ, col0, HWM, acc, nullptr, 0);
}

// ---------------- attention scores: s[z,n,m] = scale*q.k + maskbias ----------------
__global__ void __launch_bounds__(256)
k_scores(const float* __restrict__ q, const float* __restrict__ kT,
         const float* __restrict__ amb, float* __restrict__ scores, float scale)
{
    __shared__ float As[16 * ASTR];
    __shared__ float Bs[KC * BSTR];
    const int z = blockIdx.z;                  // b*NH + h
    const int b = z / NHH, h = z % NHH;
    const int row0  = blockIdx.x * 16;
    const int bcol0 = blockIdx.y * CTILE;
    const float* Qb = q  + (size_t)b * NQ * CDIM + h * HD;   // lda = CDIM, K = HD
    const float* Kb = kT + (size_t)z * HD * HW;              // ldb = HW
    float*       Sb = scores + (size_t)z * NQ * HW;
    const float* Ab = amb + (size_t)b * NQ * HW;
    v8f acc = gemm_block_tile(Qb, CDIM, row0, NQ, Kb, HW, bcol0, HW, HD, As, Bs);
    const int col0 = bcol0 + (threadIdx.x >> 5) * 16;
    const int lane = threadIdx.x & 31;
    const int half = lane >> 4;
    const int lm   = lane & 15;
    const int c    = col0 + lm;
    #pragma unroll
    for (int j = 0; j < 8; ++j) {
        const int r = row0 + j + half * 8;
        if (r < NQ)
            Sb[(size_t)r * HW + c] = acc[j] * scale + Ab[(size_t)r * HW + c];
    }
}

// ---------------- attn @ V: o[b,n,h*HD+d] = sum_m attn[z,n,m] v[b,m,h*HD+d] ----------------
__global__ void __launch_bounds__(256)
k_attnv(const float* __restrict__ attn, const float* __restrict__ v,
        float* __restrict__ o)
{
    __shared__ float As[16 * ASTR];
    __shared__ float Bs[KC * BSTR];
    const int z = blockIdx.z;
    const int b = z / NHH, h = z % NHH;
    const int row0 = blockIdx.x * 16;
    const float* Ab = attn + (size_t)z * NQ * HW;            // lda = HW, K = HW
    const float* Vb = v + (size_t)b * HW * CDIM + h * HD;    // ldb = CDIM, 32 valid cols
    float*       Ob = o + (size_t)b * NQ * CDIM + h * HD;    // ldy = CDIM
    v8f acc = gemm_block_tile(Ab, HW, row0, NQ, Vb, CDIM, 0, HD, HW, As, Bs);
    const int col0 = (threadIdx.x >> 5) * 16;
    store_tile(Ob, CDIM, row0, NQ, col0, HD, acc, nullptr, 0);
}

// ---------------- features (B,C,HW) -> feats (B,HW,C) ----------------
__global__ void k_transpose_feats(const float* __restrict__ in, float* __restrict__ out,
                                  long total)
{
    long i = (long)blockIdx.x * blockDim.x + threadIdx.x;
    if (i >= total) return;
    long b  = i / ((long)CDIM * HW);
    long r  = i % ((long)CDIM * HW);
    long c  = r / HW;
    long hw = r % HW;
    out[b * (long)HW * CDIM + hw * CDIM + c] = in[i];
}

// ---------------- kproj (B,HW,C) -> kT (B,NH,HD,HW) ----------------
__global__ void k_transpose_k(const float* __restrict__ in, float* __restrict__ out,
                              long total)
{
    long i = (long)blockIdx.x * blockDim.x + threadIdx.x;
    if (i >= total) return;
    long b = i / ((long)HW * CDIM);
    long r = i % ((long)HW * CDIM);
    long m = r / CDIM;
    long c = r % CDIM;
    long h = c / HD, d = c % HD;
    out[(((b * NHH + h) * HD) + d) * HW + m] = in[i];
}

// ---------------- broadcast query embed ----------------
__global__ void k_set_queries(const float* __restrict__ qe, float* __restrict__ q, long total)
{
    long i = (long)blockIdx.x * blockDim.x + threadIdx.x;
    if (i >= total) return;
    q[i] = qe[i % ((long)NQ * CDIM)];
}

__global__ void k_copy(const float* __restrict__ src, float* __restrict__ dst, long total)
{
    long i = (long)blockIdx.x * blockDim.x + threadIdx.x;
    if (i < total) dst[i] = src[i];
}

// ---------------- attention mask: sigmoid + 4x bilinear decimation + >0.5 + empty-row fix ----
__global__ void k_attn_mask(const float* __restrict__ cur, float* __restrict__ amb)
{
    const long row = blockIdx.x;                 // b*NQ + n
    const float* cr = cur + row * (long)HWM;
    float*       ar = amb + row * (long)HW;
    const int t = threadIdx.x;                   // 256 threads
    int pred[4]; int cnt = 0;
    #pragma unroll
    for (int i = 0; i < 4; ++i) {
        const int hw = t + i * 256;
        const int y = hw >> 5, x = hw & 31;
        const int base = (4 * y + 1) * WMQ + (4 * x + 1);
        float s = 0.f;
        s += 1.f / (1.f + __expf(-cr[base]));
        s += 1.f / (1.f + __expf(-cr[base + 1]));
        s += 1.f / (1.f + __expf(-cr[base + WMQ]));
        s += 1.f / (1.f + __expf(-cr[base + WMQ + 1]));
        pred[i] = (0.25f * s) > 0.5f;
        cnt += pred[i];
    }
    __shared__ int red[256];
    red[t] = cnt;
    __syncthreads();
    for (int s = 128; s > 0; s >>= 1) {
        if (t < s) red[t] += red[t + s];
        __syncthreads();
    }
    const int total = red[0];
    #pragma unroll
    for (int i = 0; i < 4; ++i) {
        const int hw = t + i * 256;
        ar[hw] = (total == 0 || pred[i]) ? 0.f : MASK_NEG;
    }
}

// ---------------- softmax over last dim (HW=1024), in place ----------------
__global__ void k_softmax(float* __restrict__ s)
{
    const long row = blockIdx.x;                 // (b*NH+h)*NQ + n
    float* sr = s + row * (long)HW;
    const int t = threadIdx.x;                   // 256 threads
    float v[4];
    float mx = -3.4e38f;
    #pragma unroll
    for (int i = 0; i < 4; ++i) { v[i] = sr[t + i * 256]; mx = fmaxf(mx, v[i]); }
    __shared__ float red[256];
    red[t] = mx; __syncthreads();
    for (int k = 128; k > 0; k >>= 1) {
        if (t < k) red[t] = fmaxf(red[t], red[t + k]);
        __syncthreads();
    }
    mx = red[0]; __syncthreads();
    float sum = 0.f;
    #pragma unroll
    for (int i = 0; i < 4; ++i) { v[i] = __expf(v[i] - mx); sum += v[i]; }
    red[t] = sum; __syncthreads();
    for (int k = 128; k > 0; k >>= 1) {
        if (t < k) red[t] += red[t + k];
        __syncthreads();
    }
    const float inv = 1.f / red[0];
    #pragma unroll
    for (int i = 0; i < 4; ++i) sr[t + i * 256] = v[i] * inv;
}

// ---------------- residual + layernorm (C=256, one thread per channel) ----------------
__global__ void k_residual_ln(float* __restrict__ q, const float* __restrict__ add,
                              const float* __restrict__ g, const float* __restrict__ be)
{
    const long row = blockIdx.x;                 // b*NQ + n
    const int c = threadIdx.x;                   // 256
    float* qr = q + row * (long)CDIM;
    float v = qr[c] + (add ? add[row * (long)CDIM + c] : 0.f);
    __shared__ float red[256];
    red[c] = v; __syncthreads();
    for (int k = 128; k > 0; k >>= 1) {
        if (c < k) red[c] += red[c + k];
        __syncthreads();
    }
    const float mean = red[0] * (1.f / CDIM); __syncthreads();
    const float d = v - mean;
    red[c] = d * d; __syncthreads();
    for (int k = 128; k > 0; k >>= 1) {
        if (c < k) red[c] += red[c + k];
        __syncthreads();
    }
    const float var = red[0] * (1.f / CDIM);
    qr[c] = d * rsqrtf(var + LN_EPS) * g[c] + be[c];
}

// ---------------- host-side orchestration ----------------
static inline dim3 gemm_grid(int M, int Nc, int batch)
{
    return dim3((M + 15) / 16, (Nc + CTILE - 1) / CTILE, batch);
}

extern "C" void kernel_launch(void* const* d_in, const int* in_sizes, int n_in,
                              void* d_out, int out_size, void* d_ws, size_t ws_size,
                              hipStream_t stream)
{
    const float* features      = (const float*)d_in[0];
    const float* mask_features = (const float*)d_in[1];
    const float* query_embed   = (const float*)d_in[2];
    const float* Wq  = (const float*)d_in[3];
    const float* bq  = (const float*)d_in[4];
    const float* Wk  = (const float*)d_in[5];
    const float* bk  = (const float*)d_in[6];
    const float* Wv  = (const float*)d_in[7];
    const float* bv  = (const float*)d_in[8];
    const float* Wo  = (const float*)d_in[9];
    const float* bo  = (const float*)d_in[10];
    const float* g1  = (const float*)d_in[11];
    const float* be1 = (const float*)d_in[12];
    const float* W1  = (const float*)d_in[13];
    const float* b1  = (const float*)d_in[14];
    const float* W2  = (const float*)d_in[15];
    const float* b2  = (const float*)d_in[16];
    const float* g2  = (const float*)d_in[17];
    const float* be2 = (const float*)d_in[18];
    const float* gN  = (const float*)d_in[19];
    const float* beN = (const float*)d_in[20];
    const float* Wm1 = (const float*)d_in[21];
    const float* bm1 = (const float*)d_in[22];
    const float* Wm2 = (const float*)d_in[23];
    const float* bm2 = (const float*)d_in[24];

    // output layout: queries (B,N,C), pred_masks (B,N,Hm,Wm), intermediate (L,B,N,Hm,Wm)
    float* out_q   = (float*)d_out;
    float* out_pm  = out_q + (size_t)BQ * NQ * CDIM;
    float* out_int = out_pm + (size_t)BQ * NQ * HWM;

    // workspace carve-up (floats)
    float* ws = (float*)d_ws;
    size_t o = 0;
    float* feats   = ws + o; o += (size_t)BQ * HW * CDIM;
    float* queries = ws + o; o += (size_t)BQ * NQ * CDIM;
    float* ebuf    = ws + o; o += (size_t)BQ * NQ * CDIM;
    float* tmp     = ws + o; o += (size_t)BQ * NQ * CDIM;
    float* qproj   = ws + o; o += (size_t)BQ * NQ * CDIM;
    float* kproj   = ws + o; o += (size_t)BQ * HW * CDIM;
    float* vproj   = ws + o; o += (size_t)BQ * HW * CDIM;
    float* kT      = ws + o; o += (size_t)BQ * NHH * HD * HW;
    float* scores  = ws + o; o += (size_t)BQ * NHH * NQ * HW;
    float* amb     = ws + o; o += (size_t)BQ * NQ * HW;
    float* obuf    = ws + o; o += (size_t)BQ * NQ * CDIM;

    const float scale = rsqrtf((float)HD);

    {   // feats = features.transpose
        long total = (long)BQ * CDIM * HW;
        k_transpose_feats<<<dim3((total + 255) / 256), dim3(256), 0, stream>>>(features, feats, total);
    }
    {   // broadcast queries
        long total = (long)BQ * NQ * CDIM;
        k_set_queries<<<dim3((total + 255) / 256), dim3(256), 0, stream>>>(query_embed, queries, total);
    }

    for (int l = 0; l < LNUM; ++l) {
        float* inter_l = out_int + (size_t)l * BQ * NQ * HWM;

        // predict_mask: e = relu(q@Wm1+bm1)@Wm2+bm2 ; inter_l = e @ mask_features
        k_gemm_xw<<<gemm_grid(NQ, CDIM, BQ), dim3(256), 0, stream>>>(
            queries, Wm1, bm1, tmp, NQ, CDIM, CDIM, 1);
        k_gemm_xw<<<gemm_grid(NQ, CDIM, BQ), dim3(256), 0, stream>>>(
            tmp, Wm2, bm2, ebuf, NQ, CDIM, CDIM, 0);
        k_mask_einsum<<<gemm_grid(NQ, HWM, BQ), dim3(256), 0, stream>>>(
            ebuf, mask_features, inter_l);

        // attention mask bias from sigmoid+resize+threshold
        k_attn_mask<<<dim3(BQ * NQ), dim3(256), 0, stream>>>(inter_l, amb);

        // Q/K/V projections
        k_gemm_xw<<<gemm_grid(NQ, CDIM, BQ), dim3(256), 0, stream>>>(
            queries, Wq + (size_t)l * CDIM * CDIM, bq + (size_t)l * CDIM, qproj, NQ, CDIM, CDIM, 0);
        k_gemm_xw<<<gemm_grid(HW, CDIM, BQ), dim3(256), 0, stream>>>(
            feats, Wk + (size_t)l * CDIM * CDIM, bk + (size_t)l * CDIM, kproj, HW, CDIM, CDIM, 0);
        k_gemm_xw<<<gemm_grid(HW, CDIM, BQ), dim3(256), 0, stream>>>(
            feats, Wv + (size_t)l * CDIM * CDIM, bv + (size_t)l * CDIM, vproj, HW, CDIM, CDIM, 0);
        {
            long total = (long)BQ * HW * CDIM;
            k_transpose_k<<<dim3((total + 255) / 256), dim3(256), 0, stream>>>(kproj, kT, total);
        }

        // scores + mask, softmax, attn@V
        k_scores<<<gemm_grid(NQ, HW, BQ * NHH), dim3(256), 0, stream>>>(
            qproj, kT, amb, scores, scale);
        k_softmax<<<dim3(BQ * NHH * NQ), dim3(256), 0, stream>>>(scores);
        k_attnv<<<dim3((NQ + 15) / 16, 1, BQ * NHH), dim3(256), 0, stream>>>(
            scores, vproj, obuf);

        // output projection + residual LN
        k_gemm_xw<<<gemm_grid(NQ, CDIM, BQ), dim3(256), 0, stream>>>(
            obuf, Wo + (size_t)l * CDIM * CDIM, bo + (size_t)l * CDIM, tmp, NQ, CDIM, CDIM, 0);
        k_residual_ln<<<dim3(BQ * NQ), dim3(256), 0, stream>>>(
            queries, tmp, g1 + (size_t)l * CDIM, be1 + (size_t)l * CDIM);

        // FFN (reuse scores buffer for the (B,N,F) hidden activations)
        float* ffh = scores;
        k_gemm_xw<<<gemm_grid(NQ, FDIM, BQ), dim3(256), 0, stream>>>(
            queries, W1 + (size_t)l * CDIM * FDIM, b1 + (size_t)l * FDIM, ffh, NQ, CDIM, FDIM, 1);
        k_gemm_xw<<<gemm_grid(NQ, CDIM, BQ), dim3(256), 0, stream>>>(
            ffh, W2 + (size_t)l * FDIM * CDIM, b2 + (size_t)l * CDIM, tmp, NQ, FDIM, CDIM, 0);
        k_residual_ln<<<dim3(BQ * NQ), dim3(256), 0, stream>>>(
            queries, tmp, g2 + (size_t)l * CDIM, be2 + (size_t)l * CDIM);
    }

    // final LN, export queries
    k_residual_ln<<<dim3(BQ * NQ), dim3(256), 0, stream>>>(queries, nullptr, gN, beN);
    {
        long total = (long)BQ * NQ * CDIM;
        k_copy<<<dim3((total + 255) / 256), dim3(256), 0, stream>>>(queries, out_q, total);
    }

    // final pred_masks
    k_gemm_xw<<<gemm_grid(NQ, CDIM, BQ), dim3(256), 0, stream>>>(
        queries, Wm1, bm1, tmp, NQ, CDIM, CDIM, 1);
    k_gemm_xw<<<gemm_grid(NQ, CDIM, BQ), dim3(256), 0, stream>>>(
        tmp, Wm2, bm2, ebuf, NQ, CDIM, CDIM, 0);
    k_mask_einsum<<<gemm_grid(NQ, HWM, BQ), dim3(256), 0, stream>>>(
        ebuf, mask_features, out_pm);
}